// MultiScaleFusionRoIAlign_7550552506720
// MI455X (gfx1250) — compile-verified
//
#include <hip/hip_runtime.h>
#include <hip/hip_bf16.h>
#include <cstdint>

// ---------------- problem constants (from setup_inputs) ----------------
constexpr int B_   = 2;
constexpr int C0   = 256;   // channels per pyramid level
constexpr int CT   = 768;   // fused channels
constexpr int H0   = 100, W0 = 152;   // level-0 (and fused) spatial dims
constexpr int H1   = 50,  W1 = 76;
constexpr int H2   = 25,  W2 = 38;
constexpr int NROI = 256;   // B * 128
constexpr int NS   = 14;    // OUT*SR samples per axis
constexpr int NB   = 49;    // 7*7 bins
constexpr int ROWS = 16;    // (2 samples * 2 corners)^2 gather rows per bin

// ---------------- CDNA5 helpers ----------------
#if __has_builtin(__builtin_amdgcn_s_wait_asynccnt)
#define WAIT_ASYNC(n) __builtin_amdgcn_s_wait_asynccnt(n)
#else
#define WAIT_ASYNC(n) asm volatile("s_wait_asynccnt %0" :: "i"(n) : "memory")
#endif

// one lane copies 16 contiguous bytes global -> LDS, tracked by ASYNCcnt
__device__ __forceinline__ void async_copy_b128(unsigned lds_byte_addr,
                                                unsigned long long gaddr) {
  asm volatile("global_load_async_to_lds_b128 %0, %1, off"
               :: "v"(lds_byte_addr), "v"(gaddr) : "memory");
}

// ---------------- bilinear helper (jax.image.resize, half-pixel, clamped) ----
__device__ __forceinline__ float bilin2(const float* __restrict__ base,
                                        int IH, int IW, float sy, float sx) {
  sy = fminf(fmaxf(sy, 0.f), (float)(IH - 1));
  sx = fminf(fmaxf(sx, 0.f), (float)(IW - 1));
  int iy = (int)sy, ix = (int)sx;
  float fy = sy - (float)iy, fx = sx - (float)ix;
  int iy1 = (iy + 1 < IH) ? iy + 1 : IH - 1;
  int ix1 = (ix + 1 < IW) ? ix + 1 : IW - 1;
  float v00 = base[iy * IW + ix],  v01 = base[iy * IW + ix1];
  float v10 = base[iy1 * IW + ix], v11 = base[iy1 * IW + ix1];
  return (1.f - fy) * ((1.f - fx) * v00 + fx * v01)
       +         fy * ((1.f - fx) * v10 + fx * v11);
}

// =====================================================================
// Kernel 1: fused upsample + concat + BCHW->BHWC transpose into d_ws
//   grid = (ceil(W0/32), H0, B), block = 256
// =====================================================================
__global__ __launch_bounds__(256) void fuse_kernel(
    const float* __restrict__ x0, const float* __restrict__ x1,
    const float* __restrict__ x2, float* __restrict__ feat) {
  __shared__ float tile[32][260];           // [x][c-in-level], pitch 260 (16B aligned rows)
  const int xt = blockIdx.x, y = blockIdx.y, b = blockIdx.z;
  const int lane = threadIdx.x & 31;        // x within tile
  const int row  = threadIdx.x >> 5;        // 8 channel-rows at a time
  const int xg   = xt * 32 + lane;
  const bool inb = xg < W0;

  for (int lvl = 0; lvl < 3; ++lvl) {
    // -------- phase A: read BCHW (lanes along x, coalesced) --------
    if (inb) {
      if (lvl == 0) {
        const float* p = x0 + (size_t)b * C0 * H0 * W0 + y * W0 + xg;
        for (int it = 0; it < 32; ++it) {
          int cl = row + (it << 3);
          tile[lane][cl] = p[(size_t)cl * H0 * W0];
        }
      } else if (lvl == 1) {
        const float sy = y * 0.5f - 0.25f, sx = xg * 0.5f - 0.25f;
        for (int it = 0; it < 32; ++it) {
          int cl = row + (it << 3);
          tile[lane][cl] = bilin2(x1 + (size_t)(b * C0 + cl) * H1 * W1, H1, W1, sy, sx);
        }
      } else {
        const float sy = y * 0.25f - 0.375f, sx = xg * 0.25f - 0.375f;
        for (int it = 0; it < 32; ++it) {
          int cl = row + (it << 3);
          tile[lane][cl] = bilin2(x2 + (size_t)(b * C0 + cl) * H2 * W2, H2, W2, sy, sx);
        }
      }
    }
    __syncthreads();
    // -------- phase B: write BHWC (lanes along c, float4, coalesced) --------
    const int cq = threadIdx.x & 63;        // 64 quads cover 256 channels
    for (int xi = threadIdx.x >> 6; xi < 32; xi += 4) {
      int x = xt * 32 + xi;
      if (x < W0) {
        float4 v = *(const float4*)&tile[xi][cq * 4];
        *(float4*)&feat[((size_t)((b * H0 + y) * W0) + x) * CT + lvl * C0 + cq * 4] = v;
      }
    }
    __syncthreads();
  }
}

// =====================================================================
// Per-ROI geometry: sample grid + 49x16 (row, weight) table in LDS
// =====================================================================
struct RoiShared {
  int   rowY[NB][ROWS];
  int   rowX[NB][ROWS];
  float rowW[NB][ROWS];
  int   y0[NS], y1[NS], x0[NS], x1[NS];
  float hy[NS], ly[NS], hx[NS], lx[NS], vy[NS], vx[NS];
};

__device__ void precompute_roi(RoiShared& S, int k, int t,
                               const float* __restrict__ boxes,
                               const float* __restrict__ ratio,
                               const float* __restrict__ offset,
                               const float* __restrict__ shape) {
  const int b = k >> 7;   // k / 128
  // spatial scale from padded image (batch-0 scalars, per reference)
  float padded_h = floorf(shape[0] * ratio[0] + 2.f * offset[0]);
  float padded_w = floorf(shape[1] * ratio[1] + 2.f * offset[1]);
  float ssx = (float)W0 / padded_w;
  float ssy = (float)H0 / padded_h;
  // ROI in feature coords
  float rHb = ratio[2 * b], rWb = ratio[2 * b + 1];
  float topb = offset[2 * b], leftb = offset[2 * b + 1];
  const float* bp = boxes + (size_t)k * 4;
  float rx1 = (bp[0] * rWb + leftb) * ssx;
  float ry1 = (bp[1] * rHb + topb) * ssy;
  float rx2 = (bp[2] * rWb + leftb) * ssx;
  float ry2 = (bp[3] * rHb + topb) * ssy;
  float bw = fmaxf(rx2 - rx1, 1.f) / 7.f;
  float bh = fmaxf(ry2 - ry1, 1.f) / 7.f;

  if (t < NS) {                       // y samples
    float g = ((float)t + 0.5f) * 0.5f;
    float Y = ry1 + g * bh;
    S.vy[t] = (Y >= -1.f && Y <= (float)H0) ? 1.f : 0.f;
    float Yc = fminf(fmaxf(Y, 0.f), (float)(H0 - 1));
    int yi = (int)Yc;
    S.y0[t] = yi;
    S.y1[t] = (yi + 1 < H0) ? yi + 1 : H0 - 1;
    float ly = Yc - (float)yi;
    S.ly[t] = ly; S.hy[t] = 1.f - ly;
  } else if (t < 2 * NS) {            // x samples
    int s = t - NS;
    float g = ((float)s + 0.5f) * 0.5f;
    float X = rx1 + g * bw;
    S.vx[s] = (X >= -1.f && X <= (float)W0) ? 1.f : 0.f;
    float Xc = fminf(fmaxf(X, 0.f), (float)(W0 - 1));
    int xi = (int)Xc;
    S.x0[s] = xi;
    S.x1[s] = (xi + 1 < W0) ? xi + 1 : W0 - 1;
    float lx = Xc - (float)xi;
    S.lx[s] = lx; S.hx[s] = 1.f - lx;
  }
  __syncthreads();
  // 49 bins x 16 rows: row r = (iy,ix), iy>>1 = y-sample-in-bin, iy&1 = corner
  for (int idx = t; idx < NB * ROWS; idx += blockDim.x) {
    int bin = idx >> 4, r = idx & 15;
    int by = bin / 7, bx = bin % 7;
    int iy = r >> 2, ix = r & 3;
    int sy = by * 2 + (iy >> 1), sx = bx * 2 + (ix >> 1);
    S.rowY[bin][r] = (iy & 1) ? S.y1[sy] : S.y0[sy];
    S.rowX[bin][r] = (ix & 1) ? S.x1[sx] : S.x0[sx];
    float wy = (iy & 1) ? S.ly[sy] : S.hy[sy];
    float wx = (ix & 1) ? S.lx[sx] : S.hx[sx];
    S.rowW[bin][r] = 0.25f * wy * wx * S.vy[sy] * S.vx[sx];  // 0.25 = 2x2 mean
  }
  __syncthreads();
}

// =====================================================================
// Kernel 2 (main): RoIAlign from BHWC feature map in d_ws.
// 192 threads = 6 waves per ROI; each thread owns a float4 channel group.
// Bin gathers (16 rows x 3KB, contiguous) are double-buffered in LDS via
// CDNA5 async global->LDS copies (ASYNCcnt) overlapping the FMA phase.
// =====================================================================
__global__ __launch_bounds__(192) void roialign_ws(
    const float* __restrict__ feat, const float* __restrict__ boxes,
    const float* __restrict__ ratio, const float* __restrict__ offset,
    const float* __restrict__ shape, float* __restrict__ out) {
  __shared__ RoiShared S;
  __shared__ __align__(16) float buf[2][ROWS * CT];   // 2 x 48KB staging

  const int k = blockIdx.x, t = threadIdx.x, b = k >> 7;
  precompute_roi(S, k, t, boxes, ratio, offset, shape);

  const unsigned long long featU = (unsigned long long)(uintptr_t)feat;
  const unsigned ldsBase = (unsigned)(uintptr_t)(void*)&buf[0][0];

  auto stage = [&](int bin, int slot) {
#pragma unroll
    for (int r = 0; r < ROWS; ++r) {
      unsigned long long ga = featU
          + ((unsigned long long)(unsigned)(((b * H0 + S.rowY[bin][r]) * W0
                                             + S.rowX[bin][r]) * CT) << 2)
          + (unsigned)(t * 16);
      unsigned la = ldsBase + (unsigned)(((slot * ROWS + r) * CT + t * 4) * 4);
      async_copy_b128(la, ga);   // 192 lanes x 16B = one full 3KB row
    }
  };

  stage(0, 0);
  for (int bin = 0; bin < NB; ++bin) {
    const int cur = bin & 1;
    if (bin + 1 < NB) {
      stage(bin + 1, cur ^ 1);   // prefetch next bin into other buffer
      WAIT_ASYNC(16);            // in-order: current bin's 16 copies are done
    } else {
      WAIT_ASYNC(0);
    }
    __syncthreads();             // all 6 waves' slices landed

    float4 acc = make_float4(0.f, 0.f, 0.f, 0.f);
#pragma unroll
    for (int r = 0; r < ROWS; ++r) {
      float w = S.rowW[bin][r];
      float4 v = *(const float4*)&buf[cur][r * CT + t * 4];  // ds_load_b128
      acc.x += w * v.x; acc.y += w * v.y; acc.z += w * v.z; acc.w += w * v.w;
    }
    size_t o = ((size_t)k * CT + (size_t)(t * 4)) * NB + bin;  // [K,C,7,7]
    out[o]       = acc.x;
    out[o +  NB] = acc.y;
    out[o + 2*NB]= acc.z;
    out[o + 3*NB]= acc.w;
    __syncthreads();             // buffer free before it is re-staged
  }
}

// =====================================================================
// Fallback (only if d_ws too small): direct BCHW gathers, on-the-fly fusion.
// =====================================================================
__device__ __forceinline__ float fetch_fused(const float* __restrict__ x0,
                                             const float* __restrict__ x1,
                                             const float* __restrict__ x2,
                                             int b, int y, int x, int c) {
  if (c < C0)
    return x0[(((size_t)b * C0 + c) * H0 + y) * W0 + x];
  if (c < 2 * C0)
    return bilin2(x1 + (size_t)(b * C0 + (c - C0)) * H1 * W1, H1, W1,
                  y * 0.5f - 0.25f, x * 0.5f - 0.25f);
  return bilin2(x2 + (size_t)(b * C0 + (c - 2 * C0)) * H2 * W2, H2, W2,
                y * 0.25f - 0.375f, x * 0.25f - 0.375f);
}

__global__ __launch_bounds__(192) void roialign_direct(
    const float* __restrict__ x0, const float* __restrict__ x1,
    const float* __restrict__ x2, const float* __restrict__ boxes,
    const float* __restrict__ ratio, const float* __restrict__ offset,
    const float* __restrict__ shape, float* __restrict__ out) {
  __shared__ RoiShared S;
  const int k = blockIdx.x, t = threadIdx.x, b = k >> 7;
  precompute_roi(S, k, t, boxes, ratio, offset, shape);
  const int c = t * 4;
  for (int bin = 0; bin < NB; ++bin) {
    float a0 = 0.f, a1 = 0.f, a2 = 0.f, a3 = 0.f;
    for (int r = 0; r < ROWS; ++r) {
      float w = S.rowW[bin][r];
      int Y = S.rowY[bin][r], X = S.rowX[bin][r];
      a0 += w * fetch_fused(x0, x1, x2, b, Y, X, c);
      a1 += w * fetch_fused(x0, x1, x2, b, Y, X, c + 1);
      a2 += w * fetch_fused(x0, x1, x2, b, Y, X, c + 2);
      a3 += w * fetch_fused(x0, x1, x2, b, Y, X, c + 3);
    }
    size_t o = ((size_t)k * CT + (size_t)c) * NB + bin;
    out[o] = a0; out[o + NB] = a1; out[o + 2*NB] = a2; out[o + 3*NB] = a3;
  }
}

// =====================================================================
extern "C" void kernel_launch(void* const* d_in, const int* in_sizes, int n_in,
                              void* d_out, int out_size, void* d_ws, size_t ws_size,
                              hipStream_t stream) {
  (void)in_sizes; (void)n_in; (void)out_size;
  const float* x0     = (const float*)d_in[0];
  const float* x1     = (const float*)d_in[1];
  const float* x2     = (const float*)d_in[2];
  const float* boxes  = (const float*)d_in[3];
  const float* ratio  = (const float*)d_in[4];
  const float* offset = (const float*)d_in[5];
  const float* shape  = (const float*)d_in[6];
  float* out = (float*)d_out;

  const size_t need = (size_t)B_ * H0 * W0 * CT * sizeof(float);  // 93.4 MB
  if (d_ws && ws_size >= need) {
    float* feat = (float*)d_ws;
    dim3 g1((W0 + 31) / 32, H0, B_);
    fuse_kernel<<<g1, 256, 0, stream>>>(x0, x1, x2, feat);
    roialign_ws<<<NROI, 192, 0, stream>>>(feat, boxes, ratio, offset, shape, out);
  } else {
    roialign_direct<<<NROI, 192, 0, stream>>>(x0, x1, x2, boxes, ratio, offset,
                                              shape, out);
  }
}